// WeightedMeanPreimage_8959301779819
// MI455X (gfx1250) — compile-verified
//
#include <hip/hip_runtime.h>
#include <hip/hip_bf16.h>

// WeightedMeanPreimage for MI455X (gfx1250, wave32).
// Memory-bound: stream K (512 MB, read once) at HBM rate with CDNA5 async
// global->LDS loads (ASYNCcnt, 4-buffer ring, 3 transfers in flight, NT
// cache policy so the stream doesn't evict L2-resident y), per-row top-32
// via a wave-distributed sorted list with ballot-pruned insertion, L1
// normalize, then a tiny L2-resident gather-weighted-sum of y.

#define WAVE            32
#define WAVES_PER_BLOCK 8
#define TOPK            32
#define DDIM            128
#define VEC             4   // floats per lane per stream step (b128)
#define NBUF            4   // LDS ring buffers (3 async loads in flight)
#define ASYNC_CPOL      1   // gfx12 CPol: TH[2:0]=1 -> non-temporal load

typedef float v4f __attribute__((ext_vector_type(4)));

#if defined(__has_builtin)
#  if __has_builtin(__builtin_amdgcn_global_load_async_to_lds_b128) && \
      __has_builtin(__builtin_amdgcn_s_wait_asynccnt)
#    define USE_ASYNC_LDS 1
#  endif
#endif
#ifndef USE_ASYNC_LDS
#  define USE_ASYNC_LDS 0
#endif

#if USE_ASYNC_LDS
typedef int v4i __attribute__((ext_vector_type(4)));
typedef __attribute__((address_space(1))) v4i as1_v4i;
typedef __attribute__((address_space(3))) v4i as3_v4i;
#endif

__global__ __launch_bounds__(WAVES_PER_BLOCK * WAVE)
void wmp_topk_kernel(const float* __restrict__ Kmat,
                     const float* __restrict__ Ymat,
                     float* __restrict__ outPre,
                     int*   __restrict__ outIdx,
                     int N, int B)
{
    const int lane = threadIdx.x & (WAVE - 1);
    const int wav  = threadIdx.x >> 5;
    const int row  = blockIdx.x * WAVES_PER_BLOCK + wav;

#if USE_ASYNC_LDS
    __shared__ __align__(16) float lbuf[NBUF][WAVES_PER_BLOCK][WAVE * VEC];
#endif

    if (row >= B) return;

    const float* krow  = Kmat + (size_t)row * (size_t)N;
    const int    iters = N / (WAVE * VEC);   // 32768/128 = 256

    // Distributed sorted top-32: lane l holds the (l+1)-th largest so far.
    float sval = -__builtin_inff();
    int   sidx = 0;

#if USE_ASYNC_LDS
    // Prime the ring: 3 transfers in flight.
    #pragma unroll
    for (int p = 0; p < NBUF - 1; ++p) {
        if (p < iters) {
            __builtin_amdgcn_global_load_async_to_lds_b128(
                (as1_v4i*)(krow + p * (WAVE * VEC) + lane * VEC),
                (as3_v4i*)&lbuf[p][wav][lane * VEC], 0, ASYNC_CPOL);
        }
    }
#endif

    for (int it = 0; it < iters; ++it) {
        const int base = it * (WAVE * VEC);
        v4f q;
#if USE_ASYNC_LDS
        if (it + (NBUF - 1) < iters) {
            __builtin_amdgcn_global_load_async_to_lds_b128(
                (as1_v4i*)(krow + (base + (NBUF - 1) * (WAVE * VEC)) + lane * VEC),
                (as3_v4i*)&lbuf[(it + (NBUF - 1)) % NBUF][wav][lane * VEC],
                0, ASYNC_CPOL);
            __builtin_amdgcn_s_wait_asynccnt(NBUF - 1);  // transfer `it` done
        } else {
            // tail: remaining not-yet-needed transfers = iters-1-it
            const int rem = iters - 1 - it;
            if (rem >= 2)      __builtin_amdgcn_s_wait_asynccnt(2);
            else if (rem == 1) __builtin_amdgcn_s_wait_asynccnt(1);
            else               __builtin_amdgcn_s_wait_asynccnt(0);
        }
        __asm__ __volatile__("" ::: "memory");
        q = *(const v4f*)&lbuf[it % NBUF][wav][lane * VEC];
#else
        q = __builtin_nontemporal_load(
                (const v4f*)(krow + base + lane * VEC));
#endif
        float v[VEC] = {q.x, q.y, q.z, q.w};

        // Fast skip: one ballot on the lane-max of the 4 values.
        const float thr0 = __shfl(sval, WAVE - 1);        // current 32nd best
        const float vmax = fmaxf(fmaxf(v[0], v[1]), fmaxf(v[2], v[3]));
        if (__ballot(vmax > thr0) == 0ull) continue;

        #pragma unroll
        for (int j = 0; j < VEC; ++j) {
            const float thr = __shfl(sval, WAVE - 1);
            unsigned long long m = __ballot(v[j] > thr);  // wave32: low bits
            while (m) {
                const int src = __builtin_ctzll(m);
                m &= m - 1ull;
                const float cv     = __shfl(v[j], src);
                const float curMin = __shfl(sval, WAVE - 1);
                if (cv > curMin) {   // uniform branch
                    // rank among current list (ties keep the already-held entry first)
                    const int pos =
                        __builtin_popcountll(__ballot(sval >= cv));
                    const float upv = __shfl_up(sval, 1);
                    const int   upi = __shfl_up(sidx, 1);
                    if (lane > pos)       { sval = upv; sidx = upi; }
                    else if (lane == pos) { sval = cv;  sidx = base + src * VEC + j; }
                }
            }
        }
    }

    // ---- L1 normalization of the 32 kept weights ----
    float asum = fabsf(sval);
    #pragma unroll
    for (int off = WAVE / 2; off > 0; off >>= 1)
        asum += __shfl_xor(asum, off);
    const float w = sval / fmaxf(asum, 1e-12f);

    outIdx[(size_t)row * TOPK + lane] = sidx;   // sorted descending, like top_k

    // ---- weighted sum of gathered y rows (y is L2-resident) ----
    v4f acc = (v4f)0.0f;
    #pragma unroll 4
    for (int k = 0; k < TOPK; ++k) {
        const float wk = __shfl(w, k);
        const int   ik = __shfl(sidx, k);
        const v4f yv = *(const v4f*)(Ymat + (size_t)ik * DDIM + lane * VEC);
        acc += wk * yv;
    }
    *(v4f*)(outPre + (size_t)row * DDIM + lane * VEC) = acc;
}

extern "C" void kernel_launch(void* const* d_in, const int* in_sizes, int n_in,
                              void* d_out, int out_size, void* d_ws, size_t ws_size,
                              hipStream_t stream) {
    const float* Kmat = (const float*)d_in[0];
    const float* Ymat = (const float*)d_in[1];
    // in_sizes: [B*N, N*D, 1]; D is fixed at 128 by the reference.
    const int N = in_sizes[1] / DDIM;
    const int B = in_sizes[0] / N;

    float* outPre = (float*)d_out;
    int*   outIdx = (int*)(outPre + (size_t)B * DDIM);

    dim3 block(WAVES_PER_BLOCK * WAVE);
    dim3 grid((B + WAVES_PER_BLOCK - 1) / WAVES_PER_BLOCK);
    wmp_topk_kernel<<<grid, block, 0, stream>>>(Kmat, Ymat, outPre, outIdx, N, B);
}